// DDSPHarmonicOsc_31628139167983
// MI455X (gfx1250) — compile-verified
//
#include <hip/hip_runtime.h>
#include <math.h>

#define TSAMP   64000
#define NB      32
#define NHARM   64
#define CHUNK   512
#define NCHUNK  125                 // 64000 / 512
#define TWO_PI  6.28318530717958647692f
#define LOG2E   1.44269504088896340736f
#define LN10    2.30258509299404568402f
#define INV2PI  0.15915494309189535f
#define NYQ     22050.0f            // SR/2
#define EPSV    1e-5f
#define DPHI_K  (TWO_PI / 44100.0f)

typedef float v2f __attribute__((ext_vector_type(2)));
typedef float v8f __attribute__((ext_vector_type(8)));

__device__ __forceinline__ float fexp2(float x) { return __builtin_amdgcn_exp2f(x); }
__device__ __forceinline__ float flog2(float x) { return __builtin_amdgcn_logf(x); }   // v_log_f32 (base-2)
__device__ __forceinline__ float frcp (float x) { return __builtin_amdgcn_rcpf(x);  }
__device__ __forceinline__ float fsinr(float r) { return __builtin_amdgcn_sinf(r); }   // arg in revolutions
__device__ __forceinline__ float fcosr(float r) { return __builtin_amdgcn_cosf(r); }   // arg in revolutions

// 2 * sigmoid(x)^ln(10) + 1e-7 with only 3 trans-pipe ops:
//   sigmoid(x)^ln10 = exp2(-ln10 * log2(1 + exp(-x)))   (the 1/(1+e) division cancels in the log)
__device__ __forceinline__ float scale_fn(float x) {
    float t = fexp2(-x * LOG2E);          // e^{-x}
    float u = flog2(1.0f + t);            // log2(1+e^{-x}) = -log2(sigmoid)
    float p = fexp2(-LN10 * u);           // sigmoid^ln10
    return 2.0f * p + 1e-7f;
}

// ---------------- kernel 1: per-chunk sums of dphi ----------------
__global__ void k_chunksum(const float* __restrict__ f0, float* __restrict__ csum) {
    int bc = blockIdx.x;                       // b*NCHUNK + c
    int b  = bc / NCHUNK, c = bc % NCHUNK;
    const float* f = f0 + (size_t)b * TSAMP + (size_t)c * CHUNK;
    int tid = threadIdx.x;                     // 256 threads, 2 samples each
    float v = DPHI_K * (f[2 * tid] + f[2 * tid + 1]);
    #pragma unroll
    for (int off = 16; off > 0; off >>= 1) v += __shfl_down(v, off, 32);
    __shared__ float ws[8];
    int lane = tid & 31, wid = tid >> 5;
    if (lane == 0) ws[wid] = v;
    __syncthreads();
    if (tid == 0) {
        float s = 0.f;
        #pragma unroll
        for (int i = 0; i < 8; ++i) s += ws[i];
        csum[bc] = s;
    }
}

// ---------------- kernel 2: exclusive scan of chunk sums (+phase) ----------------
__global__ void k_scan(const float* __restrict__ csum, const float* __restrict__ phase,
                       float* __restrict__ coff) {
    int b = blockIdx.x;                        // one block per batch row, 128 threads
    int i = threadIdx.x;
    __shared__ float sb[128];
    float v = (i < NCHUNK) ? csum[b * NCHUNK + i] : 0.f;
    sb[i] = v;
    __syncthreads();
    #pragma unroll
    for (int off = 1; off < 128; off <<= 1) {
        float t = (i >= off) ? sb[i - off] : 0.f;
        __syncthreads();
        sb[i] += t;
        __syncthreads();
    }
    if (i < NCHUNK) coff[b * NCHUNK + i] = phase[b] + sb[i] - v;   // exclusive + phase
}

// ---------------- kernel 3: omega scan + WMMA harmonic synthesis ----------------
__global__ void __launch_bounds__(256, 1)
k_synth(const float* __restrict__ f0, const float* __restrict__ ha,
        const float* __restrict__ coff, float* __restrict__ out) {
    int bc = blockIdx.x;
    int b  = bc / NCHUNK, c = bc % NCHUNK;
    int tid = threadIdx.x, lane = tid & 31, wid = tid >> 5;     // 8 waves of 32
    size_t rowBase = (size_t)b * TSAMP + (size_t)c * CHUNK;
    const float* fptr = f0 + rowBase;

    __shared__ float somega[CHUNK];
    __shared__ float swsum[8];

    // --- block-local inclusive scan of dphi over 512 samples (2 per thread) ---
    float d0 = DPHI_K * fptr[2 * tid];
    float d1 = DPHI_K * fptr[2 * tid + 1];
    float ps = d0 + d1;
    float s  = ps;
    #pragma unroll
    for (int off = 1; off < 32; off <<= 1) {
        float n = __shfl_up(s, off, 32);
        if (lane >= off) s += n;
    }
    if (lane == 31) swsum[wid] = s;
    __syncthreads();
    if (tid < 8) {
        float w = swsum[tid];
        #pragma unroll
        for (int off = 1; off < 8; off <<= 1) {
            float n = __shfl_up(w, off, 32);
            if (lane >= off) w += n;
        }
        swsum[tid] = w;
    }
    __syncthreads();
    float base = coff[bc] + (wid ? swsum[wid - 1] : 0.f);   // coff already includes phase
    float excl = s - ps;
    somega[2 * tid]     = base + excl + d0;
    somega[2 * tid + 1] = base + excl + d0 + d1;
    __syncthreads();

    // --- synthesis: each wave handles 4 groups of 16 samples ---
    // WMMA 16x4 f32 A-layout: lane t (t<16) covers k = 4j+{0,1}; lane t+16 covers k = 4j+{2,3}
    int hi = (lane >> 4) & 1;
    int t  = lane & 15;
    const v2f bones = {1.0f, 1.0f};            // all-ones B => D columns = row sums of A

    for (int g = 0; g < 4; ++g) {
        int   lt    = wid * 64 + g * 16 + t;   // local sample index in chunk
        size_t samp = rowBase + (size_t)lt;
        const float* xrow = ha + samp * 65ull;
        float omega = somega[lt];
        float f0v   = fptr[lt];

        if (g < 3) __builtin_prefetch(xrow + 16 * 65, 0, 0);   // global_prefetch_b8

        // pass 1: weights (each sigmoid computed exactly once per element, 3 trans each)
        float w[32];
        float lsum = 0.f;
        #pragma unroll
        for (int j = 0; j < 16; ++j) {
            int   h0 = 4 * j + 2 * hi + 1;                 // harmonic index (1..64)
            float x0 = xrow[h0];                           // element k = h0-1 of ha[1:]
            float x1 = xrow[h0 + 1];
            float a0 = scale_fn(x0);
            float a1 = scale_fn(x1);
            float w0 = (f0v * (float)h0       < NYQ) ? a0 : 0.f;
            float w1 = (f0v * (float)(h0 + 1) < NYQ) ? a1 : 0.f;
            w[2 * j]     = w0;
            w[2 * j + 1] = w1;
            lsum += w0 + w1;
        }
        float tot   = lsum + __shfl_xor(lsum, 16, 32);     // pair lanes (t, t+16) -> full sum
        float amp0  = scale_fn(xrow[0]);                   // total_amplitude
        float scale = amp0 * frcp(tot + EPSV);
        #pragma unroll
        for (int i = 0; i < 32; ++i) w[i] *= scale;

        // pass 2: sin(h*omega) via Chebyshev recurrence (step 4 per lane):
        //   sin((h+4)w) = 2cos(4w)*sin(hw) - sin((h-4)w)
        // Two interleaved chains per lane (h0 = 2*hi+1, h0+1). 5 trans ops replace 32 sins.
        float omr = omega * INV2PI;                        // revolutions
        float hb  = (float)(2 * hi + 1);
        float c4  = 2.0f * fcosr(4.0f * omr);
        float sA0 = fsinr(omr * hb);
        float sB0 = fsinr(omr * (hb + 1.0f));
        float sA1 = fsinr(omr * (hb + 4.0f));
        float sB1 = fsinr(omr * (hb + 5.0f));

        v8f cacc = {0.f, 0.f, 0.f, 0.f, 0.f, 0.f, 0.f, 0.f};
        #pragma unroll
        for (int j = 0; j < 16; ++j) {
            float sa, sb;
            if (j == 0)      { sa = sA0; sb = sB0; }
            else if (j == 1) { sa = sA1; sb = sB1; }
            else {
                sa = __builtin_fmaf(c4, sA1, -sA0);
                sb = __builtin_fmaf(c4, sB1, -sB0);
                sA0 = sA1; sA1 = sa;
                sB0 = sB1; sB1 = sb;
            }
            v2f a;
            a.x = sa * w[2 * j];
            a.y = sb * w[2 * j + 1];
            cacc = __builtin_amdgcn_wmma_f32_16x16x4_f32(
                false, a, false, bones, (short)0, cacc, false, false);
        }

        // C/D layout: lanes 0..15 hold rows 0..7 (vgpr r), lanes 16..31 rows 8..15.
        // B all-ones => every column identical; lane 0 / lane 16 store 8 results each.
        float* orow = out + rowBase + (size_t)(wid * 64 + g * 16);
        if (lane == 0) {
            #pragma unroll
            for (int i = 0; i < 8; ++i) orow[i] = cacc[i];
        } else if (lane == 16) {
            #pragma unroll
            for (int i = 0; i < 8; ++i) orow[8 + i] = cacc[i];
        }
    }
}

extern "C" void kernel_launch(void* const* d_in, const int* in_sizes, int n_in,
                              void* d_out, int out_size, void* d_ws, size_t ws_size,
                              hipStream_t stream) {
    const float* f0    = (const float*)d_in[0];   // (32, 64000)
    const float* haamp = (const float*)d_in[1];   // (32, 64000, 65)
    const float* phase = (const float*)d_in[2];   // (32,)
    float*       outp  = (float*)d_out;           // (32, 64000)

    float* csum = (float*)d_ws;                   // NB*NCHUNK
    float* coff = csum + NB * NCHUNK;             // NB*NCHUNK   (32 KB total)

    k_chunksum<<<NB * NCHUNK, 256, 0, stream>>>(f0, csum);
    k_scan    <<<NB, 128, 0, stream>>>(csum, phase, coff);
    k_synth   <<<NB * NCHUNK, 256, 0, stream>>>(f0, haamp, coff, outp);
}